// SoftStackRNNCell_old_50989851738132
// MI455X (gfx1250) — compile-verified
//
#include <hip/hip_runtime.h>
#include <hip/hip_bf16.h>

typedef __attribute__((ext_vector_type(2))) float v2f;
typedef __attribute__((ext_vector_type(8))) float v8f;

#define B_TOT      8192
#define DEPTH      512
#define SNUM       8
#define NUM_STATES 16
#define HIDDEN     128
#define EMB        64
#define NMEM       9     /* S-1+2 */
#define NBUF       4
#define FLAT       209   /* EMB + NUM_STATES + HIDDEN + 1 */
#define FLATP      212   /* padded to multiple of 4 for K loop */
#define NCOL       32    /* 29 head cols padded to 2x16 WMMA tiles */

// ---------------------------------------------------------------------------
// Kernel A: flat_input build + WMMA GEMM heads + softmax/argmax + r_new
// 2 waves per block, 16 batch rows per wave.
// ---------------------------------------------------------------------------
__global__ __launch_bounds__(64) void soft_stack_heads(
    const float* __restrict__ stack,
    const float* __restrict__ r_prev,
    const int*   __restrict__ state_prev,
    const float* __restrict__ x_emb,
    const float* __restrict__ W_r,     const float* __restrict__ b_r,
    const float* __restrict__ W_mem,   const float* __restrict__ b_mem,
    const float* __restrict__ W_buf,   const float* __restrict__ b_buf,
    const float* __restrict__ W_state, const float* __restrict__ b_state,
    float* __restrict__ out_stack, float* __restrict__ out_r,
    float* __restrict__ out_state, float* __restrict__ out_lmem,
    float* __restrict__ out_lbuf,  float* __restrict__ out_lstate,
    float4* __restrict__ ws_probs)
{
    __shared__ float sW[FLATP][NCOL];        // combined head weights, zero-padded
    __shared__ float sFlat[2][16][FLATP];    // per-wave flat_input tile
    __shared__ float sLog[2][16][NCOL];      // per-wave head logits

    const int tid   = threadIdx.x;
    const int wid   = tid >> 5;
    const int lane  = tid & 31;
    const int l16   = lane & 15;
    const int hi    = lane >> 4;        // 0: lanes 0-15, 1: lanes 16-31
    const int koff  = hi * 2;           // f32 WMMA A/B K sub-offset
    const int bbase = blockIdx.x * 32 + wid * 16;

    // ---- stage combined W = [W_mem | W_buf | W_state | 0] into LDS ----
    for (int idx = tid; idx < FLATP * NCOL; idx += 64) {
        int k = idx / NCOL, c = idx % NCOL;
        float v = 0.f;
        if (k < FLAT) {
            if (c < NMEM)                    v = W_mem[k * NMEM + c];
            else if (c < NMEM + NBUF)        v = W_buf[k * NBUF + (c - NMEM)];
            else if (c < NMEM + NBUF + NUM_STATES)
                                             v = W_state[k * NUM_STATES + (c - NMEM - NBUF)];
        }
        sW[k][c] = v;
    }

    // ---- stage static parts of flat_input for this wave's 16 rows ----
    for (int idx = lane; idx < 16 * FLATP; idx += 32) {   // 3392/32 exact, no divergence
        int row = idx / FLATP, col = idx % FLATP;
        int b = bbase + row;
        float v = 0.f;
        if (col < EMB)                       v = x_emb[b * EMB + col];
        else if (col < EMB + NUM_STATES)     v = (state_prev[b] == (col - EMB)) ? 1.f : 0.f;
        else if (col == EMB + NUM_STATES + HIDDEN)
                                             v = 1.f - stack[(size_t)b * (DEPTH * SNUM) + (SNUM - 1)];
        // r_emb region [80,208) left 0, overwritten below
        sFlat[wid][row][col] = v;
    }

    // ---- r_emb = r_prev @ W_r + b_r via v_wmma_f32_16x16x4_f32 ----
    // K=8 (2 k-steps), N=128 (8 tiles of 16)
    const int arow = bbase + l16;
    for (int nt = 0; nt < HIDDEN / 16; ++nt) {
        v8f acc = {};
        for (int ks = 0; ks < 2; ++ks) {
            int kb = ks * 4 + koff;
            v2f a, bm;
            a.x  = r_prev[arow * SNUM + kb];
            a.y  = r_prev[arow * SNUM + kb + 1];
            bm.x = W_r[kb * HIDDEN + nt * 16 + l16];
            bm.y = W_r[(kb + 1) * HIDDEN + nt * 16 + l16];
            acc = __builtin_amdgcn_wmma_f32_16x16x4_f32(
                      false, a, false, bm, (short)0, acc, false, false);
        }
        float bias = b_r[nt * 16 + l16];
        for (int i = 0; i < 8; ++i) {
            int m = i + hi * 8;   // C/D layout: VGPR i -> M=i (lo lanes) / i+8 (hi lanes)
            sFlat[wid][m][EMB + NUM_STATES + nt * 16 + l16] = acc[i] + bias;
        }
    }

    __syncthreads();

    // ---- heads GEMM: [16 x 212] @ [212 x 32] -> [16 x 32] ----
    v8f acc0 = {}, acc1 = {};
    for (int ks = 0; ks < FLATP / 4; ++ks) {
        int kb = ks * 4 + koff;
        v2f a, b0, b1;
        a.x  = sFlat[wid][l16][kb];
        a.y  = sFlat[wid][l16][kb + 1];
        b0.x = sW[kb][l16];        b0.y = sW[kb + 1][l16];
        b1.x = sW[kb][16 + l16];   b1.y = sW[kb + 1][16 + l16];
        acc0 = __builtin_amdgcn_wmma_f32_16x16x4_f32(
                   false, a, false, b0, (short)0, acc0, false, false);
        acc1 = __builtin_amdgcn_wmma_f32_16x16x4_f32(
                   false, a, false, b1, (short)0, acc1, false, false);
    }

    // ---- epilogue: bias + write logits + stage in LDS ----
    for (int t = 0; t < 2; ++t) {
        int c = t * 16 + l16;
        v8f acc = t ? acc1 : acc0;
        float bias = 0.f;
        if (c < NMEM)                          bias = b_mem[c];
        else if (c < NMEM + NBUF)              bias = b_buf[c - NMEM];
        else if (c < NMEM + NBUF + NUM_STATES) bias = b_state[c - NMEM - NBUF];
        for (int i = 0; i < 8; ++i) {
            int m = i + hi * 8;
            int b = bbase + m;
            float v = acc[i] + bias;
            sLog[wid][m][c] = v;
            if (c < NMEM)                          out_lmem[b * NMEM + c] = v;
            else if (c < NMEM + NBUF)              out_lbuf[b * NBUF + (c - NMEM)] = v;
            else if (c < NMEM + NBUF + NUM_STATES) out_lstate[b * NUM_STATES + (c - NMEM - NBUF)] = v;
        }
    }

    __syncthreads();

    // ---- per-row finalize: softmax(mem), argmax(state), r_new, probs->ws ----
    if (hi == 0) {                    // lanes 0..15 -> one batch row each
        int b = bbase + lane;
        float p[NMEM];
        float mx = -INFINITY;
        for (int j = 0; j < NMEM; ++j) { p[j] = sLog[wid][lane][j]; mx = fmaxf(mx, p[j]); }
        float sum = 0.f;
        for (int j = 0; j < NMEM; ++j) { p[j] = __expf(p[j] - mx); sum += p[j]; }
        float inv = 1.f / sum;
        for (int j = 0; j < NMEM; ++j) p[j] *= inv;
        float ppush = 0.f;
        for (int j = 0; j < SNUM - 1; ++j) ppush += p[j];
        float ppop  = p[SNUM - 1];
        float pnoop = p[SNUM];

        // argmax of the 16 state logits (first occurrence wins)
        float best = sLog[wid][lane][NMEM + NBUF];
        int   bi   = 0;
        for (int j = 1; j < NUM_STATES; ++j) {
            float v = sLog[wid][lane][NMEM + NBUF + j];
            if (v > best) { best = v; bi = j; }
        }
        out_state[b] = (float)bi;

        // r_new = new_stack[:,0] = pop*stack[b,1] + noop*stack[b,0] + push_top
        const float* st = stack + (size_t)b * (DEPTH * SNUM);
        for (int s = 0; s < SNUM; ++s) {
            float pt = (s < SNUM - 1) ? p[s] : 0.f;
            float rn = ppop * st[SNUM + s] + pnoop * st[s] + pt;
            out_r[b * SNUM + s] = rn;
            out_stack[(size_t)b * (DEPTH * SNUM) + s] = rn;
        }
        ws_probs[b] = make_float4(ppush, ppop, pnoop, 0.f);
    }
}

// ---------------------------------------------------------------------------
// Kernel B: streaming soft stack shift for d = 1..511 (memory-bound part)
// ---------------------------------------------------------------------------
__global__ __launch_bounds__(128) void soft_stack_shift(
    const float*  __restrict__ stack,
    const float4* __restrict__ ws_probs,
    float*        __restrict__ out_stack)
{
    int d = 1 + blockIdx.x * 128 + threadIdx.x;
    if (d >= DEPTH) return;
    int b = blockIdx.y;
    float4 pr = ws_probs[b];                 // {push_tot, pop, noop, -}

    const float4* cur = (const float4*)(stack + ((size_t)b * DEPTH + d)     * SNUM);
    const float4* dn  = (const float4*)(stack + ((size_t)b * DEPTH + d - 1) * SNUM);
    float4 c0 = cur[0], c1 = cur[1];
    float4 d0 = dn[0],  d1 = dn[1];
    float4 u0, u1;
    if (d < DEPTH - 1) {
        const float4* up = (const float4*)(stack + ((size_t)b * DEPTH + d + 1) * SNUM);
        u0 = up[0]; u1 = up[1];
    } else {
        u0 = make_float4(0.f, 0.f, 0.f, 0.f);
        u1 = make_float4(0.f, 0.f, 0.f, 1.f);   // null one-hot at index 7
    }
    float4 o0, o1;
    o0.x = pr.x * d0.x + pr.y * u0.x + pr.z * c0.x;
    o0.y = pr.x * d0.y + pr.y * u0.y + pr.z * c0.y;
    o0.z = pr.x * d0.z + pr.y * u0.z + pr.z * c0.z;
    o0.w = pr.x * d0.w + pr.y * u0.w + pr.z * c0.w;
    o1.x = pr.x * d1.x + pr.y * u1.x + pr.z * c1.x;
    o1.y = pr.x * d1.y + pr.y * u1.y + pr.z * c1.y;
    o1.z = pr.x * d1.z + pr.y * u1.z + pr.z * c1.z;
    o1.w = pr.x * d1.w + pr.y * u1.w + pr.z * c1.w;

    float4* out = (float4*)(out_stack + ((size_t)b * DEPTH + d) * SNUM);
    out[0] = o0;
    out[1] = o1;
}

// ---------------------------------------------------------------------------
extern "C" void kernel_launch(void* const* d_in, const int* in_sizes, int n_in,
                              void* d_out, int out_size, void* d_ws, size_t ws_size,
                              hipStream_t stream) {
    const float* stack      = (const float*)d_in[0];
    const float* r_prev     = (const float*)d_in[1];
    const int*   state_prev = (const int*)  d_in[2];
    const float* x_emb      = (const float*)d_in[3];
    // d_in[4] true_act, d_in[5] true_s, d_in[6] use_forcing: unused (forcing==0)
    const float* W_r     = (const float*)d_in[7];
    const float* b_r     = (const float*)d_in[8];
    const float* W_mem   = (const float*)d_in[9];
    const float* b_mem   = (const float*)d_in[10];
    const float* W_buf   = (const float*)d_in[11];
    const float* b_buf   = (const float*)d_in[12];
    const float* W_state = (const float*)d_in[13];
    const float* b_state = (const float*)d_in[14];

    float* out = (float*)d_out;
    size_t off = 0;
    float* out_stack  = out + off; off += (size_t)B_TOT * DEPTH * SNUM;
    float* out_r      = out + off; off += (size_t)B_TOT * SNUM;
    float* out_state  = out + off; off += (size_t)B_TOT;
    float* out_lmem   = out + off; off += (size_t)B_TOT * NMEM;
    float* out_lbuf   = out + off; off += (size_t)B_TOT * NBUF;
    float* out_lstate = out + off;

    float4* ws_probs = (float4*)d_ws;   // 8192 * 16B = 128 KB

    soft_stack_heads<<<dim3(B_TOT / 32), dim3(64), 0, stream>>>(
        stack, r_prev, state_prev, x_emb,
        W_r, b_r, W_mem, b_mem, W_buf, b_buf, W_state, b_state,
        out_stack, out_r, out_state, out_lmem, out_lbuf, out_lstate, ws_probs);

    soft_stack_shift<<<dim3((DEPTH + 127) / 128, B_TOT), dim3(128), 0, stream>>>(
        stack, ws_probs, out_stack);
}